// VoxelAggregation_77506979823776
// MI455X (gfx1250) — compile-verified
//
#include <hip/hip_runtime.h>
#include <hip/hip_bf16.h>
#include <math.h>
#include <stdint.h>

#define GRID_G 32
#define VOX    (GRID_G * GRID_G * GRID_G)   // 32768

// ---------------------------------------------------------------------------
// CDNA5 float min/max atomics, DEV scope: the RMW resolves in the 192MB L2
// (voxel grid = 67MB, fully resident) and is coherent across WGPs. HIP's
// atomicMax(float*) would lower to a CAS loop; inline asm guarantees the
// single-instruction gfx1250 path (global_atomic_max_num_f32).
// ---------------------------------------------------------------------------
__device__ __forceinline__ void atomic_max_f32_dev(float* p, float v) {
    asm volatile("global_atomic_max_num_f32 %0, %1, off scope:SCOPE_DEV"
                 :: "v"((unsigned long long)(uintptr_t)p), "v"(v)
                 : "memory");
}
__device__ __forceinline__ void atomic_min_f32_dev(float* p, float v) {
    asm volatile("global_atomic_min_num_f32 %0, %1, off scope:SCOPE_DEV"
                 :: "v"((unsigned long long)(uintptr_t)p), "v"(v)
                 : "memory");
}

// ---------------------------------------------------------------------------
// Kernel 1: init. voxel features <- -inf (scatter-max identity) with B128
// stores; counts <- 0; ws min/max <- +/-inf. All ranges divisible by 4.
// ---------------------------------------------------------------------------
__global__ void k_init(float4* __restrict__ vox4, uint4* __restrict__ cnt4,
                       float* __restrict__ wmn, float* __restrict__ wmx,
                       int totalVox4, int totalCnt4, int nMinMax) {
    const int stride = gridDim.x * blockDim.x;
    const float4 ninf4 = make_float4(-INFINITY, -INFINITY, -INFINITY, -INFINITY);
    const uint4  z4    = make_uint4(0u, 0u, 0u, 0u);
    for (int i = blockIdx.x * blockDim.x + threadIdx.x; i < totalVox4; i += stride) {
        vox4[i] = ninf4;
        if (i < totalCnt4) cnt4[i] = z4;
        if (i < nMinMax)   { wmn[i] = INFINITY; wmx[i] = -INFINITY; }
    }
}

// ---------------------------------------------------------------------------
// Kernel 2: per-batch xyz min/max. Thread-local accumulate -> LDS tree
// reduce (ds ops) -> one global float atomic per block per component.
// ---------------------------------------------------------------------------
__global__ void k_minmax(const float* __restrict__ xyz,
                         float* __restrict__ wmn, float* __restrict__ wmx,
                         int Npts) {
    const int b   = blockIdx.y;
    const int tid = threadIdx.x;
    __shared__ float smn[3][256];
    __shared__ float smx[3][256];

    float mn[3] = { INFINITY,  INFINITY,  INFINITY};
    float mx[3] = {-INFINITY, -INFINITY, -INFINITY};

    const float* base = xyz + (size_t)b * Npts * 3;
    for (int n = blockIdx.x * blockDim.x + tid; n < Npts;
         n += gridDim.x * blockDim.x) {
        const float* p = base + (size_t)n * 3;
        #pragma unroll
        for (int c = 0; c < 3; ++c) {
            float v = __builtin_nontemporal_load(p + c);
            mn[c] = fminf(mn[c], v);
            mx[c] = fmaxf(mx[c], v);
        }
    }
    #pragma unroll
    for (int c = 0; c < 3; ++c) { smn[c][tid] = mn[c]; smx[c][tid] = mx[c]; }
    __syncthreads();
    for (int s = 128; s > 0; s >>= 1) {
        if (tid < s) {
            #pragma unroll
            for (int c = 0; c < 3; ++c) {
                smn[c][tid] = fminf(smn[c][tid], smn[c][tid + s]);
                smx[c][tid] = fmaxf(smx[c][tid], smx[c][tid + s]);
            }
        }
        __syncthreads();
    }
    if (tid == 0) {
        #pragma unroll
        for (int c = 0; c < 3; ++c) {
            atomic_min_f32_dev(&wmn[b * 3 + c], smn[c][0]);
            atomic_max_f32_dev(&wmx[b * 3 + c], smx[c][0]);
        }
    }
}

// ---------------------------------------------------------------------------
// Kernel 3: voxel index per point (written as float -- values < 2^15 are
// exactly representable) + per-voxel point counts (L2-resident u32 atomics).
// ---------------------------------------------------------------------------
__global__ void k_index(const float* __restrict__ xyz,
                        const float* __restrict__ wmn,
                        const float* __restrict__ wmx,
                        float* __restrict__ out_idx,
                        unsigned* __restrict__ counts, int Npts) {
    const int b = blockIdx.y;
    const float mn0 = wmn[b * 3 + 0], mn1 = wmn[b * 3 + 1], mn2 = wmn[b * 3 + 2];
    const float r0 = wmx[b * 3 + 0] - mn0 + 1e-8f;
    const float r1 = wmx[b * 3 + 1] - mn1 + 1e-8f;
    const float r2 = wmx[b * 3 + 2] - mn2 + 1e-8f;

    for (int n = blockIdx.x * blockDim.x + threadIdx.x; n < Npts;
         n += gridDim.x * blockDim.x) {
        const float* p = xyz + ((size_t)b * Npts + n) * 3;
        float nx = fminf(fmaxf((p[0] - mn0) / r0, 0.f), 1.f - 1e-6f);
        float ny = fminf(fmaxf((p[1] - mn1) / r1, 0.f), 1.f - 1e-6f);
        float nz = fminf(fmaxf((p[2] - mn2) / r2, 0.f), 1.f - 1e-6f);
        int vx = min(max((int)floorf(nx * (float)GRID_G), 0), GRID_G - 1);
        int vy = min(max((int)floorf(ny * (float)GRID_G), 0), GRID_G - 1);
        int vz = min(max((int)floorf(nz * (float)GRID_G), 0), GRID_G - 1);
        int idx = vx * (GRID_G * GRID_G) + vy * GRID_G + vz;
        __builtin_nontemporal_store((float)idx, out_idx + (size_t)b * Npts + n);
        atomicAdd(&counts[(size_t)b * VOX + idx], 1u);
    }
}

// ---------------------------------------------------------------------------
// Kernel 4: scatter-max. One thread per point; streams D=128 feature loads
// (coalesced across lanes: consecutive lanes -> consecutive n) with TH_NT so
// the 205MB one-shot stream does NOT evict the L2-resident voxel grid, then
// fires one float-max atomic per element. Atomics are fire-and-forget
// (STOREcnt), loads are independent -> latency hidden by ~12.5k waves.
// ---------------------------------------------------------------------------
__global__ void k_scatter(const float* __restrict__ feat,
                          const float* __restrict__ out_idx,
                          float* __restrict__ vox, int Npts, int D) {
    const int b = blockIdx.y;
    for (int n = blockIdx.x * blockDim.x + threadIdx.x; n < Npts;
         n += gridDim.x * blockDim.x) {
        const int idx = (int)out_idx[(size_t)b * Npts + n];
        const float* f = feat + (size_t)b * D * Npts + n;
        float* o = vox + (size_t)b * D * VOX + idx;
        #pragma unroll 8
        for (int d = 0; d < D; ++d) {
            float v = __builtin_nontemporal_load(f + (size_t)d * Npts);
            atomic_max_f32_dev(o + (size_t)d * VOX, v);
        }
    }
}

// ---------------------------------------------------------------------------
// Kernel 5: finalize. counts -> float output; empty voxels (-inf) -> 0;
// copy per-batch min/max ([B,3,1] == flat [B*3]).
// ---------------------------------------------------------------------------
__global__ void k_finalize(const unsigned* __restrict__ counts,
                           float* __restrict__ vox,
                           float* __restrict__ out_cnt,
                           const float* __restrict__ wmn,
                           const float* __restrict__ wmx,
                           float* __restrict__ out_min,
                           float* __restrict__ out_max, int D, int BV) {
    for (int i = blockIdx.x * blockDim.x + threadIdx.x; i < BV;
         i += gridDim.x * blockDim.x) {
        unsigned c = counts[i];
        out_cnt[i] = (float)c;
        if (c == 0u) {
            int b = i / VOX, v = i - b * VOX;
            float* o = vox + (size_t)b * D * VOX + v;
            for (int d = 0; d < D; ++d) o[(size_t)d * VOX] = 0.0f;
        }
        if (i < 12) { out_min[i] = wmn[i]; out_max[i] = wmx[i]; }
    }
}

extern "C" void kernel_launch(void* const* d_in, const int* in_sizes, int n_in,
                              void* d_out, int out_size, void* d_ws, size_t ws_size,
                              hipStream_t stream) {
    const float* feat = (const float*)d_in[0];   // [B, D, N]
    const float* xyz  = (const float*)d_in[1];   // [B, N, 3]

    const int B = 4;
    const int N = in_sizes[1] / (B * 3);
    const int D = in_sizes[0] / (B * N);

    // d_out layout (flat, return order): vox[B*D*V], idx[B*N], cnt[B*V], min[B*3], max[B*3]
    float* out      = (float*)d_out;
    float* out_vox  = out;
    float* out_idx  = out_vox + (size_t)B * D * VOX;
    float* out_cnt  = out_idx + (size_t)B * N;
    float* out_min  = out_cnt + (size_t)B * VOX;
    float* out_max  = out_min + (size_t)B * 3;

    // ws layout: wmn[12], wmx[12], counts[B*V] (u32)  (~525 KB)
    float*    wmn  = (float*)d_ws;
    float*    wmx  = wmn + B * 3;
    unsigned* cnts = (unsigned*)(wmx + B * 3);

    const int totalVox4 = (B * D * VOX) / 4;   // 4,194,304
    const int totalCnt4 = (B * VOX) / 4;       // 32,768
    const int totalCnt  = B * VOX;

    k_init<<<4096, 256, 0, stream>>>((float4*)out_vox, (uint4*)cnts, wmn, wmx,
                                     totalVox4, totalCnt4, B * 3);

    dim3 gmm(256, B);
    k_minmax<<<gmm, 256, 0, stream>>>(xyz, wmn, wmx, N);

    dim3 gix((N + 255) / 256, B);
    k_index<<<gix, 256, 0, stream>>>(xyz, wmn, wmx, out_idx, cnts, N);

    dim3 gsc((N + 255) / 256, B);
    k_scatter<<<gsc, 256, 0, stream>>>(feat, out_idx, out_vox, N, D);

    k_finalize<<<(totalCnt + 255) / 256, 256, 0, stream>>>(
        cnts, out_vox, out_cnt, wmn, wmx, out_min, out_max, D, totalCnt);
}